// PyG_GCNEncoder_14130442403863
// MI455X (gfx1250) — compile-verified
//
#include <hip/hip_runtime.h>
#include <hip/hip_bf16.h>

typedef float v2f __attribute__((ext_vector_type(2)));
typedef float v8f __attribute__((ext_vector_type(8)));

#define BN_EPS 1e-5f

// ---------------------------------------------------------------------------
// Degree / normalization: deg[v] = 1 (self loop) + #in-edges; dinv = rsqrt(deg)
// ---------------------------------------------------------------------------
__global__ __launch_bounds__(256) void deg_init(float* __restrict__ deg, int n) {
    int i = blockIdx.x * blockDim.x + threadIdx.x;
    if (i < n) deg[i] = 1.0f;
}

__global__ __launch_bounds__(256) void deg_count(const long long* __restrict__ dst,
                                                 float* __restrict__ deg, int e) {
    int i = blockIdx.x * blockDim.x + threadIdx.x;
    if (i < e) atomicAdd(&deg[(int)dst[i]], 1.0f);
}

__global__ __launch_bounds__(256) void deg_finish(float* __restrict__ deg, int n) {
    int i = blockIdx.x * blockDim.x + threadIdx.x;
    if (i < n) deg[i] = rsqrtf(deg[i]);   // deg >= 1 always (self loop)
}

// ---------------------------------------------------------------------------
// Dense transform: D = A(MxK) @ W(KxNcols) [+ bias], fp32 WMMA 16x16x4.
// One wave computes a 16x64 strip (4 N-tiles): A fragment loaded once per
// K-step, reused by 4 WMMAs. M mult of 16, Ncols mult of 64, K mult of 4.
// Wave-uniform exit => EXEC all-ones for WMMA.
// ---------------------------------------------------------------------------
__global__ __launch_bounds__(256) void wmma_gemm_f32(
    const float* __restrict__ A, const float* __restrict__ W,
    const float* __restrict__ bias, float* __restrict__ D,
    int M, int K, int Ncols, int addBias)
{
    const int lane   = threadIdx.x & 31;
    const int wid    = (blockIdx.x * (blockDim.x >> 5)) + (threadIdx.x >> 5);
    const int tilesN = Ncols >> 6;               // 64-wide strips
    const int total  = (M >> 4) * tilesN;
    if (wid >= total) return;                    // uniform across the wave

    const int tm   = wid / tilesN;
    const int tn   = wid - tm * tilesN;
    const int half = lane >> 4;                  // 0: K pair {0,1}, 1: K pair {2,3}
    const int lr   = lane & 15;

    // A lane layout (ISA 32-bit A 16x4): row = lr, k = k0 + 2*half + {0,1}
    const float* ap = A + (size_t)(tm * 16 + lr) * K + (half << 1);
    // B lane layout (mirror): col = lr + 16*g, k = k0 + 2*half + {0,1}
    const float* bp = W + (size_t)(half << 1) * Ncols + tn * 64 + lr;

    v8f acc0 = {0.f,0.f,0.f,0.f,0.f,0.f,0.f,0.f};
    v8f acc1 = acc0, acc2 = acc0, acc3 = acc0;

    #pragma unroll 4
    for (int k0 = 0; k0 < K; k0 += 4) {
        v2f a = *(const v2f*)(ap + k0);          // 8B-aligned (K mult of 4)
        const float* bk = bp + (size_t)k0 * Ncols;
        v2f b0, b1, b2, b3;
        b0.x = bk[ 0];  b0.y = bk[ 0 + Ncols];
        b1.x = bk[16];  b1.y = bk[16 + Ncols];
        b2.x = bk[32];  b2.y = bk[32 + Ncols];
        b3.x = bk[48];  b3.y = bk[48 + Ncols];
        acc0 = __builtin_amdgcn_wmma_f32_16x16x4_f32(false, a, false, b0, (short)0, acc0, false, false);
        acc1 = __builtin_amdgcn_wmma_f32_16x16x4_f32(false, a, false, b1, (short)0, acc1, false, false);
        acc2 = __builtin_amdgcn_wmma_f32_16x16x4_f32(false, a, false, b2, (short)0, acc2, false, false);
        acc3 = __builtin_amdgcn_wmma_f32_16x16x4_f32(false, a, false, b3, (short)0, acc3, false, false);
    }

    // C/D layout: VGPR r -> row r + 8*half; lane lr -> col
    const int nbase = tn * 64 + lr;
    float* dp = D + (size_t)(tm * 16 + half * 8) * Ncols + nbase;
    v8f accs[4] = {acc0, acc1, acc2, acc3};
    #pragma unroll
    for (int g = 0; g < 4; ++g) {
        const float bv = addBias ? bias[nbase + g * 16] : 0.0f;
        #pragma unroll
        for (int r = 0; r < 8; ++r)
            dp[(size_t)r * Ncols + g * 16] = accs[g][r] + bv;
    }
}

// ---------------------------------------------------------------------------
// Aggregation. init: out[v,c] = dinv[v]^2 * h[v,c] + bias[c] (self-loop term)
// edges: out[dst,c] += dinv[src]*dinv[dst] * h[src,c]  (fp32 atomics, L2-resident)
// ---------------------------------------------------------------------------
__global__ __launch_bounds__(256) void agg_init(
    const float* __restrict__ h, float* __restrict__ outb,
    const float* __restrict__ dinv, const float* __restrict__ bias,
    long long total, int cshift, int cmask)
{
    long long i = (long long)blockIdx.x * blockDim.x + threadIdx.x;
    if (i >= total) return;
    int c = (int)(i & cmask);
    int v = (int)(i >> cshift);
    float d = dinv[v];
    outb[i] = d * d * h[i] + bias[c];
}

__global__ __launch_bounds__(256) void agg_edges(
    const long long* __restrict__ src, const long long* __restrict__ dst,
    const float* __restrict__ dinv, const float* __restrict__ h,
    float* __restrict__ outb, long long total, int chunkShift, int C)
{
    long long tid = (long long)blockIdx.x * blockDim.x + threadIdx.x;
    if (tid >= total) return;
    int e  = (int)(tid >> chunkShift);
    int cc = ((int)tid & ((1 << chunkShift) - 1)) << 2;  // 4 floats / thread, coalesced
    int s  = (int)src[e];
    int d  = (int)dst[e];
    float w = dinv[s] * dinv[d];
    const float4 hv = *(const float4*)(h + (size_t)s * C + cc);
    float* o = outb + (size_t)d * C + cc;
    atomicAdd(o + 0, w * hv.x);
    atomicAdd(o + 1, w * hv.y);
    atomicAdd(o + 2, w * hv.z);
    atomicAdd(o + 3, w * hv.w);
}

// ---------------------------------------------------------------------------
// BatchNorm (training-mode batch stats over node dim) + ReLU, in place.
// ---------------------------------------------------------------------------
__global__ __launch_bounds__(256) void zero_f(float* __restrict__ p, int n) {
    int i = blockIdx.x * blockDim.x + threadIdx.x;
    if (i < n) p[i] = 0.0f;
}

__global__ __launch_bounds__(256) void bn_sums(
    const float* __restrict__ x, float* __restrict__ stats,
    int N, int C, int rowsPerBlock)
{
    int r0 = blockIdx.x * rowsPerBlock;
    int r1 = r0 + rowsPerBlock; if (r1 > N) r1 = N;
    for (int c = threadIdx.x; c < C; c += blockDim.x) {
        float s = 0.0f, s2 = 0.0f;
        for (int r = r0; r < r1; ++r) {
            float v = x[(size_t)r * C + c];   // coalesced across threads
            s += v; s2 += v * v;
        }
        atomicAdd(&stats[c], s);
        atomicAdd(&stats[C + c], s2);
    }
}

__global__ __launch_bounds__(256) void bn_apply(
    float* __restrict__ x, const float* __restrict__ stats,
    const float* __restrict__ gamma, const float* __restrict__ beta,
    long long total, int cmask, int C, float invN)
{
    long long i = (long long)blockIdx.x * blockDim.x + threadIdx.x;
    if (i >= total) return;
    int c = (int)(i & cmask);
    float mu  = stats[c] * invN;
    float var = stats[C + c] * invN - mu * mu;   // biased batch variance
    float y = gamma[c] * (x[i] - mu) * rsqrtf(var + BN_EPS) + beta[c];
    x[i] = fmaxf(y, 0.0f);
}

// ---------------------------------------------------------------------------
// Orchestration
// ---------------------------------------------------------------------------
extern "C" void kernel_launch(void* const* d_in, const int* in_sizes, int n_in,
                              void* d_out, int out_size, void* d_ws, size_t ws_size,
                              hipStream_t stream) {
    const float*     x   = (const float*)d_in[0];
    const long long* ei  = (const long long*)d_in[1];
    const float*     W1  = (const float*)d_in[2];
    const float*     b1  = (const float*)d_in[3];
    const float*     g1  = (const float*)d_in[4];
    const float*     be1 = (const float*)d_in[5];
    const float*     W2  = (const float*)d_in[6];
    const float*     b2  = (const float*)d_in[7];
    const float*     g2  = (const float*)d_in[8];
    const float*     be2 = (const float*)d_in[9];
    const float*     Wl  = (const float*)d_in[10];
    const float*     bl  = (const float*)d_in[11];

    const int D_IN = 128, H1 = 256, H2 = 512, D_OUT = 128;
    const int N = in_sizes[0] / D_IN;       // 50000
    const int E = in_sizes[1] / 2;          // 500000
    const long long* srcIdx = ei;           // edge_index[0]
    const long long* dstIdx = ei + E;       // edge_index[1]

    // Workspace: dinv | stats(2*512) | hA (N*512) | hB (N*512)  ~= 205 MB
    char* ws = (char*)d_ws;
    float* dinv  = (float*)ws;
    size_t off   = (((size_t)N * 4) + 255) & ~(size_t)255;
    float* stats = (float*)(ws + off);      off += 4096;
    float* hA    = (float*)(ws + off);      off += (size_t)N * 512 * 4;
    float* hB    = (float*)(ws + off);

    const int T = 256;
    const float invN = 1.0f / (float)N;

    // --- normalization coefficients ---
    deg_init  <<<(N + T - 1) / T, T, 0, stream>>>(dinv, N);
    deg_count <<<(E + T - 1) / T, T, 0, stream>>>(dstIdx, dinv, E);
    deg_finish<<<(N + T - 1) / T, T, 0, stream>>>(dinv, N);

    // --- layer 1: GCNConv(128->256) + BN + ReLU ---
    {
        int tiles = (N >> 4) * (H1 >> 6);
        wmma_gemm_f32<<<(tiles + 7) / 8, T, 0, stream>>>(x, W1, nullptr, hA, N, D_IN, H1, 0);
        long long tot = (long long)N * H1;
        agg_init<<<(int)((tot + T - 1) / T), T, 0, stream>>>(hA, hB, dinv, b1, tot, 8, 255);
        long long etot = (long long)E << 6;                 // 64 chunks of 4 floats
        agg_edges<<<(int)((etot + T - 1) / T), T, 0, stream>>>(srcIdx, dstIdx, dinv, hA, hB, etot, 6, H1);
        zero_f<<<(2 * H1 + T - 1) / T, T, 0, stream>>>(stats, 2 * H1);
        bn_sums<<<256, T, 0, stream>>>(hB, stats, N, H1, (N + 255) / 256);
        bn_apply<<<(int)((tot + T - 1) / T), T, 0, stream>>>(hB, stats, g1, be1, tot, 255, H1, invN);
    }

    // --- layer 2: GCNConv(256->512) + BN + ReLU ---
    {
        int tiles = (N >> 4) * (H2 >> 6);
        wmma_gemm_f32<<<(tiles + 7) / 8, T, 0, stream>>>(hB, W2, nullptr, hA, N, H1, H2, 0);
        long long tot = (long long)N * H2;
        agg_init<<<(int)((tot + T - 1) / T), T, 0, stream>>>(hA, hB, dinv, b2, tot, 9, 511);
        long long etot = (long long)E << 7;                 // 128 chunks of 4 floats
        agg_edges<<<(int)((etot + T - 1) / T), T, 0, stream>>>(srcIdx, dstIdx, dinv, hA, hB, etot, 7, H2);
        zero_f<<<(2 * H2 + T - 1) / T, T, 0, stream>>>(stats, 2 * H2);
        bn_sums<<<256, T, 0, stream>>>(hB, stats, N, H2, (N + 255) / 256);
        bn_apply<<<(int)((tot + T - 1) / T), T, 0, stream>>>(hB, stats, g2, be2, tot, 511, H2, invN);
    }

    // --- final linear: out = hB @ Wl + bl ---
    {
        int tiles = (N >> 4) * (D_OUT >> 6);
        wmma_gemm_f32<<<(tiles + 7) / 8, T, 0, stream>>>(hB, Wl, bl, (float*)d_out, N, H2, D_OUT, 1);
    }
}